// GCN_78408922955888
// MI455X (gfx1250) — compile-verified
//
#include <hip/hip_runtime.h>
#include <hip/hip_bf16.h>
#include <math.h>

#define N_NODES 50000

typedef __attribute__((ext_vector_type(2))) float v2f;
typedef __attribute__((ext_vector_type(8))) float v8f;

// ---------------------------------------------------------------------------
// zero fill (float4 stores; n is a multiple of 4 for all our buffers)
// ---------------------------------------------------------------------------
__global__ void gcn_zero_f32(float4* __restrict__ p, int n4) {
    int i = blockIdx.x * blockDim.x + threadIdx.x;
    int stride = gridDim.x * blockDim.x;
    float4 z = {0.f, 0.f, 0.f, 0.f};
    for (; i < n4; i += stride) p[i] = z;
}

// ---------------------------------------------------------------------------
// scatter-add aggregation: out[dst[e], :] += x[src[e], :]
// one wave per edge; lane handles 4 consecutive floats (global_load_b128 =>
// the whole source row is one contiguous wave request). fp32 HW atomics.
// Working set is L2-resident so the atomic RMWs resolve at L2 bandwidth.
// ---------------------------------------------------------------------------
__global__ void gcn_scatter_add(const float* __restrict__ x,
                                const long long* __restrict__ src,
                                const long long* __restrict__ dst,
                                float* __restrict__ out, int E, int F) {
    int e = (int)((blockIdx.x * (unsigned)blockDim.x + threadIdx.x) >> 5);
    int lane = threadIdx.x & 31;
    if (e >= E) return;
    int s = __builtin_amdgcn_readfirstlane((int)src[e]);
    int d = __builtin_amdgcn_readfirstlane((int)dst[e]);
    const float* xs = x + s * F;
    float*       od = out + d * F;
    for (int f = lane * 4; f < F; f += 128) {
        float4 v = *(const float4*)(xs + f);
        unsafeAtomicAdd(&od[f + 0], v.x);
        unsafeAtomicAdd(&od[f + 1], v.y);
        unsafeAtomicAdd(&od[f + 2], v.z);
        unsafeAtomicAdd(&od[f + 3], v.w);
    }
}

// ---------------------------------------------------------------------------
// C[M x N] = A[M x K] @ W[K x N] + bias, optional relu.
// Block = 16-row tile of A (staged in LDS, padded stride). One wave per
// 16-col output tile. fp32 WMMA 16x16x4 per the CDNA5 ISA layouts:
//   A frag : lanes 0-15 -> M=lane,  v0=K(k0), v1=K(k0+1)
//            lanes16-31 -> M=l-16,  v0=K(k0+2), v1=K(k0+3)
//   B frag : lanes 0-15 -> N=lane,  v0=B[k0][n],   v1=B[k0+1][n]
//            lanes16-31 -> N=l-16,  v0=B[k0+2][n], v1=B[k0+3][n]
//   C/D    : vgpr r, lane l -> M = r + 8*(l>=16), N = l%16
// K,N compile-time: fully unrolled loop -> LDS dword-imm loads, global loads
// with immediate offsets, back-to-back v_wmma. Out-of-range columns (N=40
// case) are CLAMPED on load (values are garbage but never stored) so the
// inner loop has no EXEC divergence. All index math is 32-bit.
// ---------------------------------------------------------------------------
template <int K, int N, bool RELU>
__global__ void gcn_gemm_bias_act(const float* __restrict__ A,
                                  const float* __restrict__ W,
                                  const float* __restrict__ bias,
                                  float* __restrict__ out) {
    constexpr int LDK = K + 1;  // pad to avoid LDS bank conflicts on row reads
    __shared__ float lds[16 * LDK];
    const int m0 = blockIdx.x * 16;

    // cooperative stage of the 16xK A tile (no runtime division)
    {
        const float* ap = A + m0 * K;
        for (int r = 0; r < 16; ++r)
            for (int c = threadIdx.x; c < K; c += blockDim.x)
                lds[r * LDK + c] = ap[r * K + c];
    }
    __syncthreads();

    const int wave = threadIdx.x >> 5;
    const int lane = threadIdx.x & 31;
    const int half = lane >> 4;       // 0: lanes 0-15, 1: lanes 16-31
    const int lp   = lane & 15;
    const int n0   = wave * 16;
    const int n    = n0 + lp;
    const int nc   = (n < N) ? n : (N - 1);  // clamp: garbage cols never stored

    const float* ap = lds + lp * LDK + 2 * half;  // A-frag base (per lane)
    const float* wp = W + 2 * half * N + nc;      // B-frag base (per lane)

    v8f acc = {};
#pragma unroll
    for (int k0 = 0; k0 < K; k0 += 4) {
        v2f a, b;
        a.x = ap[k0];
        a.y = ap[k0 + 1];
        b.x = wp[k0 * N];
        b.y = wp[k0 * N + N];
        acc = __builtin_amdgcn_wmma_f32_16x16x4_f32(
            /*neg_a=*/false, a, /*neg_b=*/false, b,
            /*c_mod=*/(short)0, acc, /*reuse_a=*/false, /*reuse_b=*/false);
    }

    if (n >= N) return;
    const float bv = bias[n];
    float* op = out + (m0 + 8 * half) * N + n;
#pragma unroll
    for (int r = 0; r < 8; ++r) {
        float v = acc[r] + bv;
        if (RELU) v = fmaxf(v, 0.0f);
        op[r * N] = v;  // lanes write 16 consecutive floats per r
    }
}

// ---------------------------------------------------------------------------
// row-wise log_softmax, N <= 64. One wave per row; wave32 shuffle reductions.
// ---------------------------------------------------------------------------
__global__ void gcn_log_softmax(const float* __restrict__ in,
                                float* __restrict__ out, int M, int N) {
    int row  = (int)((blockIdx.x * (unsigned)blockDim.x + threadIdx.x) >> 5);
    int lane = threadIdx.x & 31;
    if (row >= M) return;
    const float* r = in + row * N;
    float v0 = (lane      < N) ? r[lane]      : -INFINITY;
    float v1 = (lane + 32 < N) ? r[lane + 32] : -INFINITY;
    float m = fmaxf(v0, v1);
#pragma unroll
    for (int off = 16; off > 0; off >>= 1)
        m = fmaxf(m, __shfl_xor(m, off, 32));
    float s = 0.0f;
    if (lane      < N) s += expf(v0 - m);
    if (lane + 32 < N) s += expf(v1 - m);
#pragma unroll
    for (int off = 16; off > 0; off >>= 1)
        s += __shfl_xor(s, off, 32);
    float lse = m + logf(s);
    float* o = out + row * N;
    if (lane      < N) o[lane]      = v0 - lse;
    if (lane + 32 < N) o[lane + 32] = v1 - lse;
}

// ---------------------------------------------------------------------------
// launch
// ---------------------------------------------------------------------------
extern "C" void kernel_launch(void* const* d_in, const int* in_sizes, int n_in,
                              void* d_out, int out_size, void* d_ws, size_t ws_size,
                              hipStream_t stream) {
    const float*     x   = (const float*)d_in[0];
    const float*     W1  = (const float*)d_in[1];
    const float*     b1  = (const float*)d_in[2];
    const float*     W2  = (const float*)d_in[3];
    const float*     b2  = (const float*)d_in[4];
    const float*     W3  = (const float*)d_in[5];
    const float*     b3  = (const float*)d_in[6];
    const long long* ei  = (const long long*)d_in[7];
    const int E = in_sizes[7] / 2;
    const long long* src = ei;
    const long long* dst = ei + E;
    float* out = (float*)d_out;

    // workspace: ping-pong buffers
    float* bufA = (float*)d_ws;                        // 50000*128 f32 (25.6 MB)
    float* bufB = bufA + (size_t)N_NODES * 128;        // 50000*96  f32 (19.2 MB)

    const int ZB = 2048, ZT = 256;
    const int scatterBlocks = (E + 7) / 8;             // 8 waves (edges) / block
    const int gemmBlocks = N_NODES / 16;               // 3125, exact

    // ---- layer 1: agg(x) -> A ; relu(A@W1+b1) -> B -------------------------
    gcn_zero_f32<<<ZB, ZT, 0, stream>>>((float4*)bufA, N_NODES * 128 / 4);
    gcn_scatter_add<<<scatterBlocks, 256, 0, stream>>>(x, src, dst, bufA, E, 128);
    gcn_gemm_bias_act<128, 96, true><<<gemmBlocks, 6 * 32, 0, stream>>>(bufA, W1, b1, bufB);

    // ---- layer 2: agg(B) -> A ; relu(A@W2+b2) -> B -------------------------
    gcn_zero_f32<<<ZB, ZT, 0, stream>>>((float4*)bufA, N_NODES * 96 / 4);
    gcn_scatter_add<<<scatterBlocks, 256, 0, stream>>>(bufB, src, dst, bufA, E, 96);
    gcn_gemm_bias_act<96, 96, true><<<gemmBlocks, 6 * 32, 0, stream>>>(bufA, W2, b2, bufB);

    // ---- layer 3: agg(B) -> A ; A@W3+b3 -> B(stride 40) ; log_softmax ------
    gcn_zero_f32<<<ZB, ZT, 0, stream>>>((float4*)bufA, N_NODES * 96 / 4);
    gcn_scatter_add<<<scatterBlocks, 256, 0, stream>>>(bufB, src, dst, bufA, E, 96);
    gcn_gemm_bias_act<96, 40, false><<<gemmBlocks, 3 * 32, 0, stream>>>(bufA, W3, b3, bufB);
    gcn_log_softmax<<<(N_NODES + 7) / 8, 256, 0, stream>>>(bufB, out, N_NODES, 40);
}